// DAGCN_5652176961768
// MI455X (gfx1250) — compile-verified
//
#include <hip/hip_runtime.h>
#include <hip/hip_bf16.h>
#include <stdint.h>

#define N_NODES 150002
#define DIM     128
#define N_EDGES 600000
#define EPSN    1e-12f
#define ROWSTRIDE 132   // 128 + 4 pad -> conflict-free strided LDS reads

typedef float __attribute__((ext_vector_type(2))) v2f;
typedef float __attribute__((ext_vector_type(8))) v8f;

__device__ __forceinline__ float4 ld4(const float* p) { return *reinterpret_cast<const float4*>(p); }
__device__ __forceinline__ void   st4(float* p, float4 v) { *reinterpret_cast<float4*>(p) = v; }

// ---------------------------------------------------------------------------
// COO SpMM: y[row] += (val*mask) * x[col].  One edge per wave32, lane handles
// 4 consecutive floats. x (76.8MB) is L2-resident; atomics resolve in L2.
// ---------------------------------------------------------------------------
__global__ void __launch_bounds__(256)
spmm_kernel(const int* __restrict__ rows, const int* __restrict__ cols,
            const float* __restrict__ vals, const uint8_t* __restrict__ mask,
            const float* __restrict__ x, float* __restrict__ y)
{
    const int lane = threadIdx.x & 31;
    const int wv   = (blockIdx.x * blockDim.x + threadIdx.x) >> 5;
    const int nw   = (gridDim.x * blockDim.x) >> 5;
    for (int e = wv; e < N_EDGES; e += nw) {
        // prefetch next iteration's gather row (global_prefetch_b8)
        const int en = e + nw;
        if (en < N_EDGES && mask[en]) {
            const int cn = cols[en];
            __builtin_prefetch(x + (size_t)cn * DIM + lane * 4, 0, 0);
        }
        if (!mask[e]) continue;              // uniform per wave: no divergence
        const int   c  = cols[e];
        const int   r  = rows[e];
        const float w  = vals[e];
        const float4 xv = ld4(x + (size_t)c * DIM + lane * 4);
        float* yp = y + (size_t)r * DIM + lane * 4;
        atomicAdd(yp + 0, w * xv.x);
        atomicAdd(yp + 1, w * xv.y);
        atomicAdd(yp + 2, w * xv.z);
        atomicAdd(yp + 3, w * xv.w);
    }
}

// ---------------------------------------------------------------------------
// o = (a + b + c) / 3, elementwise float4
// ---------------------------------------------------------------------------
__global__ void __launch_bounds__(256)
avg3_kernel(const float* __restrict__ a, const float* __restrict__ b,
            const float* __restrict__ c, float* __restrict__ o)
{
    const size_t n4 = (size_t)N_NODES * (DIM / 4);
    const size_t i  = (size_t)blockIdx.x * blockDim.x + threadIdx.x;
    if (i >= n4) return;
    const float s = 1.0f / 3.0f;
    const float4 va = ld4(a + i * 4), vb = ld4(b + i * 4), vc = ld4(c + i * 4);
    float4 vo;
    vo.x = s * (va.x + vb.x + vc.x);
    vo.y = s * (va.y + vb.y + vc.y);
    vo.z = s * (va.z + vb.z + vc.z);
    vo.w = s * (va.w + vb.w + vc.w);
    st4(o + i * 4, vo);
}

// ---------------------------------------------------------------------------
// Per-wave row-norm via WMMA: LDS holds agg^2 tile [16 x 132]. Accumulate
// 32x V_WMMA_F32_16X16X4_F32 with B = ones: D[m][n] = sum_k agg[m][k]^2
// replicated over n.  A layout (16x4 f32): lanes 0-15 hold {K0,K1} of row
// M=lane, lanes 16-31 hold {K2,K3} of row M=lane-16.
// Extract: norm2[i] = VGPR(i&7) of lane (i>>3)*16.
// ---------------------------------------------------------------------------
__device__ __forceinline__ void wave_rownorm_inv(const float* ldsw, int lane, float* invn)
{
    v8f c;
    #pragma unroll
    for (int j = 0; j < 8; ++j) c[j] = 0.0f;
    v2f bones; bones[0] = 1.0f; bones[1] = 1.0f;
    const int rsel = lane & 15;
    const int hi   = lane >> 4;
    #pragma unroll
    for (int k = 0; k < 32; ++k) {
        const float* p = ldsw + rsel * ROWSTRIDE + 4 * k + 2 * hi;   // 8B aligned
        v2f a = *reinterpret_cast<const v2f*>(p);
        c = __builtin_amdgcn_wmma_f32_16x16x4_f32(
                /*neg_a=*/false, a, /*neg_b=*/false, bones,
                /*c_mod=*/(short)0, c, /*reuse_a=*/false, /*reuse_b=*/false);
    }
    #pragma unroll
    for (int i = 0; i < 16; ++i) {
        const float n2 = __shfl(c[i & 7], (i >> 3) << 4, 32);
        invn[i] = 1.0f / fmaxf(sqrtf(n2), EPSN);
    }
}

// cur_aux = l2norm(trans * (emb + t0 + t1)/3) + emb
__global__ void __launch_bounds__(128)
finalize_aux_kernel(const float* __restrict__ emb, const float* __restrict__ t0,
                    const float* __restrict__ t1, const float* __restrict__ trans,
                    float* __restrict__ out)
{
    __shared__ float lds[4][16 * ROWSTRIDE];
    const int lane = threadIdx.x & 31;
    const int wv   = threadIdx.x >> 5;
    const int row0 = blockIdx.x * 64 + wv * 16;
    float4 aggv[16];
    #pragma unroll
    for (int i = 0; i < 16; ++i) {
        const int rc = min(row0 + i, N_NODES - 1);   // clamp: keep EXEC all-1s
        const size_t off = (size_t)rc * DIM + lane * 4;
        const float4 e = ld4(emb + off), a = ld4(t0 + off), b = ld4(t1 + off);
        const float s = trans[rc] * (1.0f / 3.0f);
        float4 g;
        g.x = s * (e.x + a.x + b.x);
        g.y = s * (e.y + a.y + b.y);
        g.z = s * (e.z + a.z + b.z);
        g.w = s * (e.w + a.w + b.w);
        aggv[i] = g;
        float* d = &lds[wv][i * ROWSTRIDE + lane * 4];
        d[0] = g.x * g.x; d[1] = g.y * g.y; d[2] = g.z * g.z; d[3] = g.w * g.w;
    }
    __syncthreads();
    float invn[16];
    wave_rownorm_inv(lds[wv], lane, invn);
    #pragma unroll
    for (int i = 0; i < 16; ++i) {
        const int r = row0 + i;
        if (r >= N_NODES) continue;
        const size_t off = (size_t)r * DIM + lane * 4;
        const float4 e = ld4(emb + off);
        const float4 g = aggv[i];
        float4 o;
        o.x = g.x * invn[i] + e.x;
        o.y = g.y * invn[i] + e.y;
        o.z = g.z * invn[i] + e.z;
        o.w = g.w * invn[i] + e.w;
        st4(out + off, o);
    }
}

// out = l2norm(tball*gall + tbaux*(cur+t0+t1)/3) + cur   (gall aliases out)
__global__ void __launch_bounds__(128)
finalize_buy_kernel(const float* __restrict__ cur, const float* __restrict__ t0,
                    const float* __restrict__ t1, const float* __restrict__ gall,
                    const float* __restrict__ tball, const float* __restrict__ tbaux,
                    float* __restrict__ out)
{
    __shared__ float lds[4][16 * ROWSTRIDE];
    const int lane = threadIdx.x & 31;
    const int wv   = threadIdx.x >> 5;
    const int row0 = blockIdx.x * 64 + wv * 16;
    float4 aggv[16];
    #pragma unroll
    for (int i = 0; i < 16; ++i) {
        const int rc = min(row0 + i, N_NODES - 1);
        const size_t off = (size_t)rc * DIM + lane * 4;
        const float4 ca = ld4(cur + off), a = ld4(t0 + off), b = ld4(t1 + off);
        const float4 ga = ld4(gall + off);
        const float s1 = tball[rc];
        const float s2 = tbaux[rc] * (1.0f / 3.0f);
        float4 g;
        g.x = s1 * ga.x + s2 * (ca.x + a.x + b.x);
        g.y = s1 * ga.y + s2 * (ca.y + a.y + b.y);
        g.z = s1 * ga.z + s2 * (ca.z + a.z + b.z);
        g.w = s1 * ga.w + s2 * (ca.w + a.w + b.w);
        aggv[i] = g;
        float* d = &lds[wv][i * ROWSTRIDE + lane * 4];
        d[0] = g.x * g.x; d[1] = g.y * g.y; d[2] = g.z * g.z; d[3] = g.w * g.w;
    }
    __syncthreads();
    float invn[16];
    wave_rownorm_inv(lds[wv], lane, invn);
    #pragma unroll
    for (int i = 0; i < 16; ++i) {
        const int r = row0 + i;
        if (r >= N_NODES) continue;
        const size_t off = (size_t)r * DIM + lane * 4;
        const float4 ca = ld4(cur + off);
        const float4 g  = aggv[i];
        float4 o;
        o.x = g.x * invn[i] + ca.x;
        o.y = g.y * invn[i] + ca.y;
        o.z = g.z * invn[i] + ca.z;
        o.w = g.w * invn[i] + ca.w;
        st4(out + off, o);
    }
}

// ---------------------------------------------------------------------------
extern "C" void kernel_launch(void* const* d_in, const int* in_sizes, int n_in,
                              void* d_out, int out_size, void* d_ws, size_t ws_size,
                              hipStream_t stream)
{
    const float*   emb      = (const float*)d_in[0];
    const int*     rows_aux = (const int*)d_in[1];
    const int*     cols_aux = (const int*)d_in[2];
    const float*   vals_aux = (const float*)d_in[3];
    const int*     rows_buy = (const int*)d_in[4];
    const int*     cols_buy = (const int*)d_in[5];
    const float*   vals_buy = (const float*)d_in[6];
    const float*   tr_aux_all = (const float*)d_in[7];
    const float*   tr_buy_all = (const float*)d_in[8];
    const float*   tr_buy_aux = (const float*)d_in[9];
    const uint8_t* masks    = (const uint8_t*)d_in[10];   // (6, E) bool

    const size_t ND = (size_t)N_NODES * DIM;
    float* t0  = (float*)d_ws;
    float* t1  = t0 + ND;
    float* cur = t1 + ND;
    float* gall = (float*)d_out;

    const int SPMM_B = 4096, SPMM_T = 256;                // 32768 waves
    const int FIN_B  = (N_NODES + 63) / 64;
    const int AVG_T  = 256;
    const int AVG_B  = (int)(((size_t)N_NODES * (DIM / 4) + AVG_T - 1) / AVG_T);

    // --- behavior 'aux': g = gcn(emb, A_aux) ; cur = l2norm(tr*g) + emb ---
    hipMemsetAsync(t0, 0, ND * sizeof(float), stream);
    spmm_kernel<<<SPMM_B, SPMM_T, 0, stream>>>(rows_aux, cols_aux, vals_aux,
                                               masks + 0 * (size_t)N_EDGES, emb, t0);
    hipMemsetAsync(t1, 0, ND * sizeof(float), stream);
    spmm_kernel<<<SPMM_B, SPMM_T, 0, stream>>>(rows_aux, cols_aux, vals_aux,
                                               masks + 1 * (size_t)N_EDGES, t0, t1);
    finalize_aux_kernel<<<FIN_B, 128, 0, stream>>>(emb, t0, t1, tr_aux_all, cur);

    // --- behavior 'buy', pre 'all': g_all = gcn(emb, A_buy) -> d_out ---
    hipMemsetAsync(t0, 0, ND * sizeof(float), stream);
    spmm_kernel<<<SPMM_B, SPMM_T, 0, stream>>>(rows_buy, cols_buy, vals_buy,
                                               masks + 2 * (size_t)N_EDGES, emb, t0);
    hipMemsetAsync(t1, 0, ND * sizeof(float), stream);
    spmm_kernel<<<SPMM_B, SPMM_T, 0, stream>>>(rows_buy, cols_buy, vals_buy,
                                               masks + 3 * (size_t)N_EDGES, t0, t1);
    avg3_kernel<<<AVG_B, AVG_T, 0, stream>>>(emb, t0, t1, gall);

    // --- behavior 'buy', pre 'aux': g_aux = gcn(cur, A_buy) ---
    hipMemsetAsync(t0, 0, ND * sizeof(float), stream);
    spmm_kernel<<<SPMM_B, SPMM_T, 0, stream>>>(rows_buy, cols_buy, vals_buy,
                                               masks + 4 * (size_t)N_EDGES, cur, t0);
    hipMemsetAsync(t1, 0, ND * sizeof(float), stream);
    spmm_kernel<<<SPMM_B, SPMM_T, 0, stream>>>(rows_buy, cols_buy, vals_buy,
                                               masks + 5 * (size_t)N_EDGES, t0, t1);

    // --- out = l2norm(tball*g_all + tbaux*g_aux) + cur ---
    finalize_buy_kernel<<<FIN_B, 128, 0, stream>>>(cur, t0, t1, gall,
                                                   tr_buy_all, tr_buy_aux, (float*)d_out);
}